// DenseGATLayer_88914412962523
// MI455X (gfx1250) — compile-verified
//
#include <hip/hip_runtime.h>
#include <hip/hip_bf16.h>
#include <math.h>

typedef __attribute__((ext_vector_type(2))) float v2f;
typedef __attribute__((ext_vector_type(8))) float v8f;

#define N_ROWS   4096
#define IN_DIM   1024
#define OUT_DIM  512
#define N_HEADS  4
#define HEAD_DIM 128
#define TOPK     16
#define NEG_SLOPE 0.2f

// ---------------------------------------------------------------------------
// Kernel 1: Wh = x @ W.T   (fp32 WMMA, V_WMMA_F32_16X16X4_F32)
// Register-blocked: each wave computes a 32x32 output tile with 4 v8f
// accumulators -> per K-step: 2 A-frag + 2 B-frag loads feed 4 WMMAs
// (1:1 load:wmma, each fragment reused twice from registers).
// Block = 128 threads (4 waves) covering 32 rows x 128 cols.
// grid = (M/32, OUT_DIM/128)
// ---------------------------------------------------------------------------
__global__ __launch_bounds__(128)
void gemm_wmma_f32(const float* __restrict__ X, const float* __restrict__ W,
                   float* __restrict__ Wh) {
    const int lane = threadIdx.x & 31;
    const int wave = threadIdx.x >> 5;
    const int m0   = blockIdx.x * 32;
    const int n0   = blockIdx.y * 128 + wave * 32;

    const int half = lane >> 4;        // 0 = lanes 0-15, 1 = lanes 16-31
    const int sub  = lane & 15;
    const int kb   = half * 2;         // K sub-offset per ISA A/B fragment layout

    // A fragments: rows m0+sub and m0+16+sub, K pair {kb, kb+1}  (b64 loads)
    const float* Aptr0 = X + (size_t)(m0 + sub)      * IN_DIM + kb;
    const float* Aptr1 = X + (size_t)(m0 + 16 + sub) * IN_DIM + kb;
    // B fragments (B = W^T): output cols n0+sub and n0+16+sub
    const float* Bptr0 = W + (size_t)(n0 + sub)      * IN_DIM + kb;
    const float* Bptr1 = W + (size_t)(n0 + 16 + sub) * IN_DIM + kb;

    v8f acc00 = {}, acc01 = {}, acc10 = {}, acc11 = {};

    #pragma unroll 4
    for (int k = 0; k < IN_DIM; k += 4) {
        const v2f a0 = *(const v2f*)(Aptr0 + k);
        const v2f a1 = *(const v2f*)(Aptr1 + k);
        const v2f b0 = *(const v2f*)(Bptr0 + k);
        const v2f b1 = *(const v2f*)(Bptr1 + k);
        acc00 = __builtin_amdgcn_wmma_f32_16x16x4_f32(false, a0, false, b0,
                                                      (short)0, acc00, false, false);
        acc01 = __builtin_amdgcn_wmma_f32_16x16x4_f32(false, a0, false, b1,
                                                      (short)0, acc01, false, false);
        acc10 = __builtin_amdgcn_wmma_f32_16x16x4_f32(false, a1, false, b0,
                                                      (short)0, acc10, false, false);
        acc11 = __builtin_amdgcn_wmma_f32_16x16x4_f32(false, a1, false, b1,
                                                      (short)0, acc11, false, false);
    }

    // C/D layout: VGPR r -> row (tile_m + r + 8*half), col (tile_n + sub)
    {
        float* Cp = Wh + (size_t)(m0 + half * 8) * OUT_DIM + (n0 + sub);
        #pragma unroll
        for (int r = 0; r < 8; ++r) Cp[(size_t)r * OUT_DIM] = acc00[r];
        Cp += 16;   // +16 cols
        #pragma unroll
        for (int r = 0; r < 8; ++r) Cp[(size_t)r * OUT_DIM] = acc01[r];
    }
    {
        float* Cp = Wh + (size_t)(m0 + 16 + half * 8) * OUT_DIM + (n0 + sub);
        #pragma unroll
        for (int r = 0; r < 8; ++r) Cp[(size_t)r * OUT_DIM] = acc10[r];
        Cp += 16;
        #pragma unroll
        for (int r = 0; r < 8; ++r) Cp[(size_t)r * OUT_DIM] = acc11[r];
    }
}

// ---------------------------------------------------------------------------
// Kernel 2: s_src[n][h] = dot(Wh[n, h*128:(h+1)*128], a[:128])
//           s_dst[n][h] = dot(Wh[n, h*128:(h+1)*128], a[128:256])
// one thread per (n, h)
// ---------------------------------------------------------------------------
__global__ __launch_bounds__(128)
void attn_scores(const float* __restrict__ Wh, const float* __restrict__ a,
                 float* __restrict__ s_src, float* __restrict__ s_dst) {
    const int t = blockIdx.x * blockDim.x + threadIdx.x;
    if (t >= N_ROWS * N_HEADS) return;
    const int n = t >> 2;
    const int h = t & 3;
    const float* p  = Wh + (size_t)n * OUT_DIM + h * HEAD_DIM;
    const float* as = a;
    const float* ad = a + HEAD_DIM;
    float ss = 0.f, sd = 0.f;
    #pragma unroll 4
    for (int d = 0; d < HEAD_DIM; ++d) {
        const float v = p[d];
        ss += v * as[d];
        sd += v * ad[d];
    }
    s_src[t] = ss;
    s_dst[t] = sd;
}

// ---------------------------------------------------------------------------
// Kernel 3: per output row i:
//   top-16 of adj[i,:] (iterative argmax in LDS, lower-index tie-break),
//   leaky-relu scores, per-head softmax over the 16 neighbors,
//   out[i,h,:] = sum_j alpha[h][j] * Wh[nbr_j, h,:], then ELU.
// one 256-thread block per row
// ---------------------------------------------------------------------------
__global__ __launch_bounds__(256)
void gat_row(const float* __restrict__ adj, const float* __restrict__ Wh,
             const float* __restrict__ s_src, const float* __restrict__ s_dst,
             float* __restrict__ out) {
    __shared__ float row[N_ROWS];
    __shared__ float redv[256];
    __shared__ int   redi[256];
    __shared__ int   nbr[TOPK];
    __shared__ float alpha[N_HEADS][TOPK];

    const int i = blockIdx.x;
    const int t = threadIdx.x;

    // stage adj row in LDS (16 KB of the 320 KB WGP LDS)
    for (int c = t; c < N_ROWS; c += 256)
        row[c] = adj[(size_t)i * N_ROWS + c];
    __syncthreads();

    // 16 successive argmax extractions
    for (int sel = 0; sel < TOPK; ++sel) {
        float best = -INFINITY;
        int   bi   = 0x7fffffff;
        for (int c = t; c < N_ROWS; c += 256) {
            const float v = row[c];
            if (v > best || (v == best && c < bi)) { best = v; bi = c; }
        }
        redv[t] = best; redi[t] = bi;
        __syncthreads();
        for (int s = 128; s > 0; s >>= 1) {
            if (t < s) {
                const float v2 = redv[t + s];
                const int   i2 = redi[t + s];
                if (v2 > redv[t] || (v2 == redv[t] && i2 < redi[t])) {
                    redv[t] = v2; redi[t] = i2;
                }
            }
            __syncthreads();
        }
        if (t == 0) {
            nbr[sel] = redi[0];
            row[redi[0]] = -INFINITY;   // remove from next round
        }
        __syncthreads();
    }

    // per-head softmax over the 16 selected neighbors (threads 0..3)
    if (t < N_HEADS) {
        const int h = t;
        const float si = s_src[i * N_HEADS + h];
        float e[TOPK];
        float mx = -INFINITY;
        #pragma unroll
        for (int j = 0; j < TOPK; ++j) {
            float v = si + s_dst[nbr[j] * N_HEADS + h];
            v = v > 0.f ? v : NEG_SLOPE * v;        // leaky relu
            e[j] = v;
            mx = fmaxf(mx, v);
        }
        float sum = 0.f;
        #pragma unroll
        for (int j = 0; j < TOPK; ++j) {
            const float ex = expf(e[j] - mx);
            alpha[h][j] = ex;
            sum += ex;
        }
        const float inv = 1.0f / sum;
        #pragma unroll
        for (int j = 0; j < TOPK; ++j) alpha[h][j] *= inv;
    }
    __syncthreads();

    // aggregate + ELU: 512 output dims, 2 per thread; gathers hit L2-resident Wh
    for (int o = t; o < OUT_DIM; o += 256) {
        const int h = o >> 7;   // o / HEAD_DIM
        float acc = 0.f;
        #pragma unroll
        for (int j = 0; j < TOPK; ++j)
            acc += alpha[h][j] * Wh[(size_t)nbr[j] * OUT_DIM + o];
        out[(size_t)i * OUT_DIM + o] = acc > 0.f ? acc : (expf(acc) - 1.0f);
    }
}

// ---------------------------------------------------------------------------
extern "C" void kernel_launch(void* const* d_in, const int* in_sizes, int n_in,
                              void* d_out, int out_size, void* d_ws, size_t ws_size,
                              hipStream_t stream) {
    const float* x   = (const float*)d_in[0];   // (4096, 1024)
    const float* adj = (const float*)d_in[1];   // (4096, 4096)
    const float* W   = (const float*)d_in[2];   // (512, 1024)
    const float* a   = (const float*)d_in[3];   // (1, 256)
    float* out = (float*)d_out;                 // (4096, 512)

    float* Wh    = (float*)d_ws;                        // 4096*512 floats (8 MB)
    float* s_src = Wh + (size_t)N_ROWS * OUT_DIM;       // 4096*4
    float* s_dst = s_src + (size_t)N_ROWS * N_HEADS;    // 4096*4

    // Wh = x @ W.T  via fp32 WMMA, 32x32 register-blocked tiles per wave
    dim3 gg(N_ROWS / 32, OUT_DIM / 128);
    gemm_wmma_f32<<<gg, 128, 0, stream>>>(x, W, Wh);

    // attention source/dest scores
    const int nt = N_ROWS * N_HEADS;
    attn_scores<<<(nt + 127) / 128, 128, 0, stream>>>(Wh, a, s_src, s_dst);

    // per-row top-k + softmax + aggregation + ELU
    gat_row<<<N_ROWS, 256, 0, stream>>>(adj, Wh, s_src, s_dst, out);
}